// VectorQuantizer_1924145349236
// MI455X (gfx1250) — compile-verified
//
#include <hip/hip_runtime.h>

// ---- CDNA5 (gfx1250) VQ-VAE vector quantizer ----
// fp32 WMMA (V_WMMA_F32_16X16X4_F32) fused distance-GEMM + argmin,
// with TDM (tensor_load_to_lds) double-buffered codebook staging.

typedef float v2f __attribute__((ext_vector_type(2)));
typedef float v8f __attribute__((ext_vector_type(8)));
typedef unsigned int v4u __attribute__((ext_vector_type(4)));
typedef int v4i __attribute__((ext_vector_type(4)));
typedef int v8i __attribute__((ext_vector_type(8)));

#define K_CODES 1024
#define DIMS    256
#define NPIX    32768          // 32 * 32 * 32 pixels
#define M_TILE  64             // pixels per workgroup
#define LSTR    260            // LDS row stride (256 + 4 pad -> conflict-free b64 reads)
#define TOTAL_ELEMS 8388608    // 32*256*32*32

// ---------------- kernel 1: code norms + zero histogram ----------------
__global__ __launch_bounds__(256) void vq_prep(const float* __restrict__ cb,
                                               float* __restrict__ codeNorms,
                                               int* __restrict__ counts) {
  const int k = blockIdx.x;      // one code per block
  const int t = threadIdx.x;     // 256 threads
  __shared__ float s[256];
  float v = cb[k * DIMS + t];
  s[t] = v * v;
  __syncthreads();
  #pragma unroll
  for (int st = 128; st > 0; st >>= 1) {
    if (t < st) s[t] += s[t + st];
    __syncthreads();
  }
  if (t == 0) {
    codeNorms[k] = s[0];
    counts[k] = 0;               // re-zero every launch (graph-replay safe)
  }
}

__device__ inline void halfwave_argmin(float& dv, int& iv) {
  #pragma unroll
  for (int off = 1; off < 16; off <<= 1) {
    float od = __shfl_xor(dv, off, 32);
    int   oi = __shfl_xor(iv, off, 32);
    if (od < dv || (od == dv && oi < iv)) { dv = od; iv = oi; }
  }
}

// ---- TDM: DMA one 64x256 fp32 codebook tile (row-contiguous) into LDS,
//      inserting 4 DWORDs of padding every 256 DWORDs (-> 260-float rows).
__device__ inline void tdm_load_tile(const float* gsrc, unsigned int lds_byte_addr) {
  unsigned long long ga = (unsigned long long)(size_t)gsrc;
  v4u g0;
  g0.x = 1u;                                     // count = 1 valid descriptor
  g0.y = lds_byte_addr;                          // LDS destination (bytes)
  g0.z = (unsigned int)ga;                       // global_addr[31:0]
  g0.w = (unsigned int)(ga >> 32) | (2u << 30);  // global_addr[56:32] | type=2
  v8i g1;
  g1[0] = (int)((2u << 16)      // data_size = 4 bytes
              | (1u << 20)      // pad_enable
              | (7u << 22)      // pad_interval: 256 DWORDs (one 1 KB row)
              | (3u << 25));    // pad_amount:   4 DWORDs (16 B)
  g1[1] = (int)(256u << 16);    // tensor_dim0 = 256 (bits 79:48)
  g1[2] = (int)(1024u << 16);   // tensor_dim1 = 1024 (bits 111:80)
  g1[3] = (int)(256u << 16);    // tile_dim0 = 256 (bits 127:112)
  g1[4] = 64;                   // tile_dim1 = 64, tile_dim2 = 0
  g1[5] = 256;                  // tensor_dim0_stride = 256 (bits 207:160)
  g1[6] = 0;                    // stride hi / tensor_dim1_stride = 0
  g1[7] = 0;
  v4i z4 = {0, 0, 0, 0};                         // groups 2/3 unused (2-D tile)
  v8i z8 = {0, 0, 0, 0, 0, 0, 0, 0};             // trailing group (unused)
  __builtin_amdgcn_tensor_load_to_lds(g0, g1, z4, z4, z8, 0);
}

// ---------------- kernel 2: fused distance GEMM + argmin + gather ----------------
__global__ __launch_bounds__(256) void vq_main(const float* __restrict__ x,
                                               const float* __restrict__ cb,
                                               const float* __restrict__ codeNorms,
                                               float* __restrict__ out,
                                               int* __restrict__ counts,
                                               float* __restrict__ blockSSE) {
  __shared__ float ldsA[M_TILE * LSTR];   // 64 pixels x 256 dims (padded)
  __shared__ float ldsB[2][64 * LSTR];    // double-buffered 64-code tiles (TDM dest)
  __shared__ float redD[64][4];
  __shared__ int   redI[64][4];
  __shared__ int   finalIdx[64];
  __shared__ float ldsRed[256];

  const int t    = threadIdx.x;
  const int g    = blockIdx.x;       // 512 blocks, 64 pixels each
  const int lane = t & 31;
  const int w    = t >> 5;           // 8 waves
  const int nSub = w & 3;            // 4 N-subtiles of 16 codes
  const int mGrp = w >> 2;           // 2 M-groups, 2 tiles (32 pixels) each
  const int b    = g >> 4;           // image index  (16 tiles of 64 pix per image)
  const int hw0  = (g & 15) * M_TILE;
  const int xbase = (b << 18) + hw0; // b*C*H*W + hw0

  // ---- TDM prefetch of codebook chunk 0 (overlaps with the A-tile staging) ----
  if (w == 0) {
    tdm_load_tile(cb, (unsigned int)(size_t)&ldsB[0][0]);
  }

  // ---- load 64x256 pixel tile (transpose from [B,C,H,W]) ----
  for (int i = t; i < M_TILE * DIMS; i += 256) {
    int p = i & 63;
    int d = i >> 6;
    ldsA[p * LSTR + d] = x[xbase + (d << 10) + p];   // coalesced along p
  }

  const int l15   = lane & 15;
  const int koff  = 2 * (lane >> 4);                 // WMMA f32 A/B K-half select
  const int rowA0 = (mGrp * 32 + l15) * LSTR;
  const int rowA1 = rowA0 + 16 * LSTR;
  const int rowB  = (nSub * 16 + l15) * LSTR;

  float bestD0[8], bestD1[8];
  int   bestI0[8], bestI1[8];
  #pragma unroll
  for (int v = 0; v < 8; ++v) {
    bestD0[v] = 3.4e38f; bestD1[v] = 3.4e38f;
    bestI0[v] = 0;       bestI1[v] = 0;
  }

  for (int chunk = 0; chunk < 16; ++chunk) {
    const int buf = chunk & 1;
    // prefetch next chunk into the other buffer, then wait for this chunk's DMA
    if (w == 0) {
      if (chunk + 1 < 16) {
        tdm_load_tile(cb + (chunk + 1) * 64 * DIMS,
                      (unsigned int)(size_t)&ldsB[buf ^ 1][0]);
        __builtin_amdgcn_s_wait_tensorcnt(1);   // chunk `chunk` complete
      } else {
        __builtin_amdgcn_s_wait_tensorcnt(0);   // last chunk complete
      }
    }
    __syncthreads();   // publish TDM-written tile (and A tile on first pass)

    const float* Bb = &ldsB[buf][0];
    const int   nLoc = chunk * 64 + nSub * 16 + l15;
    const float nb   = codeNorms[nLoc];          // L2-resident, hidden under WMMA

    v8f acc0 = {0.f, 0.f, 0.f, 0.f, 0.f, 0.f, 0.f, 0.f};
    v8f acc1 = {0.f, 0.f, 0.f, 0.f, 0.f, 0.f, 0.f, 0.f};

    #pragma unroll 8
    for (int k0 = 0; k0 < DIMS; k0 += 4) {
      v2f bb = *(const v2f*)(Bb   + rowB  + k0 + koff);
      v2f a0 = *(const v2f*)(ldsA + rowA0 + k0 + koff);
      v2f a1 = *(const v2f*)(ldsA + rowA1 + k0 + koff);
      // D = A(16x4,f32) * B(4x16,f32) + C   -> v_wmma_f32_16x16x4_f32
      acc0 = __builtin_amdgcn_wmma_f32_16x16x4_f32(false, a0, false, bb,
                                                   (short)0, acc0, false, false);
      acc1 = __builtin_amdgcn_wmma_f32_16x16x4_f32(false, a1, false, bb,
                                                   (short)0, acc1, false, false);
    }

    // dist = ||c||^2 - 2*x.c   (||x||^2 constant per row -> dropped)
    #pragma unroll
    for (int v = 0; v < 8; ++v) {
      float d0 = nb - 2.0f * acc0[v];
      if (d0 < bestD0[v]) { bestD0[v] = d0; bestI0[v] = nLoc; }
      float d1 = nb - 2.0f * acc1[v];
      if (d1 < bestD1[v]) { bestD1[v] = d1; bestI1[v] = nLoc; }
    }
    __syncthreads();   // all reads of buf done before TDM(chunk+2) overwrites it
  }

  // ---- reduce argmin across the 16 lanes of each half-wave, publish per-wave bests ----
  #pragma unroll
  for (int v = 0; v < 8; ++v) {
    float dv = bestD0[v]; int iv = bestI0[v];
    halfwave_argmin(dv, iv);
    if (lane == 0)  { redD[mGrp * 32 + v][nSub]      = dv; redI[mGrp * 32 + v][nSub]      = iv; }
    if (lane == 16) { redD[mGrp * 32 + 8 + v][nSub]  = dv; redI[mGrp * 32 + 8 + v][nSub]  = iv; }

    dv = bestD1[v]; iv = bestI1[v];
    halfwave_argmin(dv, iv);
    if (lane == 0)  { redD[mGrp * 32 + 16 + v][nSub] = dv; redI[mGrp * 32 + 16 + v][nSub] = iv; }
    if (lane == 16) { redD[mGrp * 32 + 24 + v][nSub] = dv; redI[mGrp * 32 + 24 + v][nSub] = iv; }
  }
  __syncthreads();

  // ---- combine the 4 N-subtile candidates per pixel; histogram ----
  if (t < 64) {
    float bd = redD[t][0]; int bi = redI[t][0];
    #pragma unroll
    for (int s = 1; s < 4; ++s) {
      float dd = redD[t][s]; int ii = redI[t][s];
      if (dd < bd || (dd == bd && ii < bi)) { bd = dd; bi = ii; }
    }
    finalIdx[t] = bi;
    atomicAdd(&counts[bi], 1);
  }
  __syncthreads();

  // ---- gather quantized codes, write output [B,C,H,W], accumulate SSE ----
  const int p  = t & 63;
  const int dg = t >> 6;         // 4 dim-groups of 64
  const int k  = finalIdx[p];
  float sse = 0.0f;
  #pragma unroll 4
  for (int j = 0; j < 64; ++j) {
    int d = dg * 64 + j;
    float q  = cb[k * DIMS + d];           // L2-resident scattered gather
    float xv = ldsA[p * LSTR + d];
    out[xbase + (d << 10) + p] = q;        // coalesced along p
    float df = q - xv;
    sse += df * df;
  }

  // deterministic per-block SSE partial
  ldsRed[t] = sse;
  __syncthreads();
  #pragma unroll
  for (int s = 128; s > 0; s >>= 1) {
    if (t < s) ldsRed[t] += ldsRed[t + s];
    __syncthreads();
  }
  if (t == 0) blockSSE[g] = ldsRed[0];
}

// ---------------- kernel 3: loss + perplexity (deterministic reduction) ----------------
__global__ __launch_bounds__(1024) void vq_final(const float* __restrict__ blockSSE,
                                                 const int* __restrict__ counts,
                                                 float* __restrict__ out) {
  const int t = threadIdx.x;     // 1024 threads
  __shared__ float sa[1024];
  __shared__ float sb[1024];
  sa[t] = (t < 512) ? blockSSE[t] : 0.0f;
  float pr = (float)counts[t] * (1.0f / (float)NPIX);
  sb[t] = pr * logf(pr + 1e-10f);
  __syncthreads();
  #pragma unroll
  for (int s = 512; s > 0; s >>= 1) {
    if (t < s) { sa[t] += sa[t + s]; sb[t] += sb[t + s]; }
    __syncthreads();
  }
  if (t == 0) {
    // loss = q_latent + 0.25*e_latent = 1.25 * SSE / numel
    out[TOTAL_ELEMS]     = 1.25f * sa[0] / (float)TOTAL_ELEMS;
    out[TOTAL_ELEMS + 1] = expf(-sb[0]);
  }
}

extern "C" void kernel_launch(void* const* d_in, const int* in_sizes, int n_in,
                              void* d_out, int out_size, void* d_ws, size_t ws_size,
                              hipStream_t stream) {
  const float* x  = (const float*)d_in[0];   // [32,256,32,32] fp32
  const float* cb = (const float*)d_in[1];   // [1024,256] fp32
  float* out = (float*)d_out;                // [8388608] out + loss + perplexity

  float* wsf       = (float*)d_ws;
  float* codeNorms = wsf;                    // [1024] f32
  float* blockSSE  = wsf + 1024;             // [512]  f32
  int*   counts    = (int*)(wsf + 2048);     // [1024] i32

  vq_prep <<<K_CODES, 256, 0, stream>>>(cb, codeNorms, counts);
  vq_main <<<NPIX / M_TILE, 256, 0, stream>>>(x, cb, codeNorms, out, counts, blockSSE);
  vq_final<<<1, 1024, 0, stream>>>(blockSSE, counts, out);
}